// WaveRNN_88141318848589
// MI455X (gfx1250) — compile-verified
//
#include <hip/hip_runtime.h>
#include <stdint.h>

// Problem constants (match reference)
#define B_   4
#define NZ_  256
#define NX_  256
#define NT_  256
#define NR_  128
#define ROWS 8                       // grid rows per block
#define NBLK ((B_ * NZ_) / ROWS)     // 128 blocks, 32 per shot
#define DT_        1e-3f
#define DH_        10.0f
#define TAU_SIGMA_ 0.1f              // 1/OMEGA

typedef unsigned int uint32;
typedef uint32 v4u __attribute__((ext_vector_type(4)));
typedef int    v8i __attribute__((ext_vector_type(8)));
typedef int    v4i __attribute__((ext_vector_type(4)));

// Generic pointer to LDS: low 32 bits of the flat address are the LDS offset
__device__ __forceinline__ uint32 lds_offset(const void* p) {
    return (uint32)(uint64_t)(uintptr_t)p;
}

// 1-D contiguous Tensor-Data-Mover load: global -> LDS, ndw dwords (<= 65535).
// D# built per CDNA5 ISA (group0/group1 bitfields).
// This toolchain's builtin takes 6 args: (g0 v4u, g1 v8i, g2 v4i, g3 v4i, v8i, i32 cpol).
__device__ __forceinline__ void tdm_load_1d(uint32 lds, const void* gsrc, uint32 ndw) {
    uint64_t ga = (uint64_t)(uintptr_t)gsrc;
    v4u g0;
    g0[0] = 1u;                                           // count=1 (valid), not restore, no gather
    g0[1] = lds;                                          // lds_addr (bytes)
    g0[2] = (uint32)(ga & 0xffffffffull);                 // global_addr[31:0]
    g0[3] = (uint32)((ga >> 32) & 0x1ffffffull)           // global_addr[56:32]
          | (2u << 30);                                   // type = 2 ("image")
    v8i g1;
    g1[0] = (int)(2u << 16);                              // wg_mask=0 (not in cluster), data_size=2 (4B)
    g1[1] = (int)((ndw & 0xffffu) << 16);                 // tensor_dim0[15:0]
    g1[2] = (int)(((ndw >> 16) & 0xffffu) | (1u << 16));  // tensor_dim0[31:16], tensor_dim1=1
    g1[3] = (int)((ndw & 0xffffu) << 16);                 // tensor_dim1[31:16]=0, tile_dim0=ndw
    g1[4] = 0;                                            // tile_dim1=0 (unused -> 1D), tile_dim2=0
    g1[5] = (int)ndw;                                     // tensor_dim0_stride[31:0]
    g1[6] = 0;                                            // stride0[47:32], stride1[15:0]
    g1[7] = 0;                                            // stride1[47:16]
    v4i gz4 = {0, 0, 0, 0};                               // groups 2/3 unused (2-D descriptor)
    v8i gz8 = {0, 0, 0, 0, 0, 0, 0, 0};
    __builtin_amdgcn_tensor_load_to_lds(g0, g1, gz4, gz4, gz8, 0 /*cpol*/);
}

// Grid-wide software barrier: monotonic counter in scratch, reset each launch.
__device__ __forceinline__ void grid_barrier(uint32* bar, uint32 target) {
    __threadfence();                 // make halo stores device-visible before arrive
    __syncthreads();
    if (threadIdx.x == 0) {
        __hip_atomic_fetch_add(bar, 1u, __ATOMIC_ACQ_REL, __HIP_MEMORY_SCOPE_AGENT);
        while (__hip_atomic_load(bar, __ATOMIC_ACQUIRE, __HIP_MEMORY_SCOPE_AGENT) < target) {
            __builtin_amdgcn_s_sleep(1);
        }
    }
    __syncthreads();
}

__global__ __launch_bounds__(NX_, 1)
void wave_persistent(const float* __restrict__ wav,   // [B,NT] source wavelet
                     const float* __restrict__ vp,    // [NZ,NX]
                     const float* __restrict__ rho,   // [NZ,NX]
                     const float* __restrict__ Q,     // [NZ,NX]
                     const int*   __restrict__ srcZ,  // [B]
                     const int*   __restrict__ srcX,  // [B]
                     const int*   __restrict__ recX,  // [NR]
                     const int*   __restrict__ recZp, // [1]
                     float*       __restrict__ out,   // [NT,NR,B]
                     float*       __restrict__ pHalo, // [NBLK][NX]  (row j=7 of each block)
                     float*       __restrict__ vzHalo,// [NBLK][NX]  (row j=0 of each block)
                     uint32*      __restrict__ bar)
{
    // All four wavefields + coefficients live entirely in LDS (~58 KB/block).
    __shared__ float sP [ROWS][NX_];
    __shared__ float sVx[ROWS][NX_];
    __shared__ float sVz[ROWS][NX_];
    __shared__ float sR [ROWS][NX_];
    __shared__ float sCv[ROWS][NX_];   // DT/(rho*DH)
    __shared__ float sCb[ROWS][NX_];   // (DT/tau_sigma)*kappa*tau/(cb*DH)
    __shared__ float sCc[ROWS][NX_];   // DT*kappa*(1+tau)/DH
    __shared__ float sXi[NT_];
    __shared__ int   sRecX[NR_];

    const int x  = threadIdx.x;              // column (0..255)
    const int bb = blockIdx.x;               // block id (0..127)
    const int b  = bb >> 5;                  // shot (32 blocks per shot)
    const int z0 = (bb & 31) * ROWS;         // first grid row of this block

    // ---- TDM bulk DMA of parameters into LDS (wave 0 issues; EXEC all-ones) ----
    if ((threadIdx.x >> 5) == 0) {
        tdm_load_1d(lds_offset(&sVx[0][0]), vp  + z0 * NX_, ROWS * NX_); // stage vp
        tdm_load_1d(lds_offset(&sVz[0][0]), rho + z0 * NX_, ROWS * NX_); // stage rho
        tdm_load_1d(lds_offset(&sP [0][0]), Q   + z0 * NX_, ROWS * NX_); // stage Q
        tdm_load_1d(lds_offset(&sXi[0]),    wav + b  * NT_, NT_);
        tdm_load_1d(lds_offset(&sRecX[0]),  recX,           NR_);
        __builtin_amdgcn_s_wait_tensorcnt(0);
    }
    __syncthreads();

    // ---- Precompute per-cell coefficients from staged params ----
    const float Acb = 0.995f / 1.005f;       // ca/cb
    #pragma unroll
    for (int j = 0; j < ROWS; ++j) {
        float v   = sVx[j][x];               // vp
        float rh  = sVz[j][x];               // rho
        float q   = sP [j][x];               // Q
        float kap = rh * v * v;
        float tau = 1.0f / q;
        sCv[j][x] = DT_ / (rh * DH_);
        sCb[j][x] = (DT_ / TAU_SIGMA_) * kap * tau / (1.005f * DH_);
        sCc[j][x] = DT_ * kap * (1.0f + tau) / DH_;
    }
    __syncthreads();
    #pragma unroll
    for (int j = 0; j < ROWS; ++j) {
        sP[j][x] = 0.0f; sVx[j][x] = 0.0f; sVz[j][x] = 0.0f; sR[j][x] = 0.0f;
    }
    // Zero our own halo slots (visible to neighbors after first grid barrier)
    __hip_atomic_store(&pHalo [bb * NX_ + x], 0.0f, __ATOMIC_RELEASE, __HIP_MEMORY_SCOPE_AGENT);
    __hip_atomic_store(&vzHalo[bb * NX_ + x], 0.0f, __ATOMIC_RELEASE, __HIP_MEMORY_SCOPE_AGENT);

    const int  sz      = srcZ[b];
    const int  sx      = srcX[b];
    const int  rz      = recZp[0];
    const bool srcHere = (sz >= z0) && (sz < z0 + ROWS) && (x == sx);
    const int  srcJ    = sz - z0;
    const bool recBlk  = (rz >= z0) && (rz < z0 + ROWS);
    const int  recJ    = rz - z0;
    const bool topShot = (z0 == 0);                 // z==0 : backward-diff pad
    const bool botShot = ((bb & 31) == 31);         // z==NZ-1 : forward-diff pad

    const float* pPrev  = pHalo  + (bb - 1) * NX_;  // p(z0-1) row, owned by bb-1
    const float* vzNext = vzHalo + (bb + 1) * NX_;  // vz(z0+8) row, owned by bb+1

    uint32 k = 0;
    const uint32 nb = gridDim.x;

    for (int t = 0; t < NT_; ++t) {
        // ---- barrier: previous step's p halos (or initial zeros) are visible ----
        grid_barrier(bar, (++k) * nb);

        float pzm = topShot ? 0.0f
            : __hip_atomic_load(&pPrev[x], __ATOMIC_ACQUIRE, __HIP_MEMORY_SCOPE_AGENT);

        // ---- velocity update (reads p, writes vx/vz; all LDS) ----
        float vz0 = 0.0f;
        #pragma unroll
        for (int j = 0; j < ROWS; ++j) {
            float pc   = sP[j][x];
            float pxm  = (x == 0) ? 0.0f : sP[j][x - 1];
            float pzmj = (j == 0) ? pzm  : sP[j - 1][x];
            float cv   = sCv[j][x];
            float nvx  = sVx[j][x] - cv * (pc - pxm);
            float nvz  = sVz[j][x] - cv * (pc - pzmj);
            sVx[j][x] = nvx;
            sVz[j][x] = nvz;
            if (j == 0) vz0 = nvz;
        }
        // publish top vz row for block bb-1
        __hip_atomic_store(&vzHalo[bb * NX_ + x], vz0, __ATOMIC_RELEASE, __HIP_MEMORY_SCOPE_AGENT);

        grid_barrier(bar, (++k) * nb);

        float vzBot = botShot ? 0.0f
            : __hip_atomic_load(&vzNext[x], __ATOMIC_ACQUIRE, __HIP_MEMORY_SCOPE_AGENT);

        // ---- stress/memory-variable update + source injection ----
        float xi = sXi[t];
        float p7 = 0.0f;
        #pragma unroll
        for (int j = 0; j < ROWS; ++j) {
            float vxc  = sVx[j][x];
            float vxp  = (x == NX_ - 1) ? 0.0f : sVx[j][x + 1];
            float vzc  = sVz[j][x];
            float vzp  = (j == ROWS - 1) ? vzBot : sVz[j + 1][x];
            float divh = (vxp - vxc) + (vzp - vzc);        // DH folded into coeffs
            float rn   = Acb * sR[j][x] - sCb[j][x] * divh;
            sR[j][x]   = rn;
            float pn   = sP[j][x] - sCc[j][x] * divh + DT_ * rn;
            if (srcHere && j == srcJ) pn += xi;            // source injection
            sP[j][x]   = pn;
            if (j == ROWS - 1) p7 = pn;
        }
        // publish bottom p row for block bb+1
        __hip_atomic_store(&pHalo[bb * NX_ + x], p7, __ATOMIC_RELEASE, __HIP_MEMORY_SCOPE_AGENT);

        // ---- receiver readout (block-local; source row == receiver row case safe) ----
        __syncthreads();
        if (recBlk && x < NR_) {
            out[(t * NR_ + x) * B_ + b] = sP[recJ][sRecX[x]];
        }
    }
}

extern "C" void kernel_launch(void* const* d_in, const int* in_sizes, int n_in,
                              void* d_out, int out_size, void* d_ws, size_t ws_size,
                              hipStream_t stream) {
    const float* wav  = (const float*)d_in[0];   // x  [B,NT,1]
    const float* vp   = (const float*)d_in[1];   // [NZ,NX]
    const float* rho  = (const float*)d_in[2];   // [NZ,NX]
    const float* Q    = (const float*)d_in[3];   // [NZ,NX]
    const int*   srcZ = (const int*)d_in[4];     // [B]
    const int*   srcX = (const int*)d_in[5];     // [B]
    const int*   recX = (const int*)d_in[6];     // [NR]
    const int*   recZ = (const int*)d_in[7];     // [1]
    float*       out  = (float*)d_out;           // [NT,NR,B]

    // Scratch layout: p halos (128KB) | vz halos (128KB) | barrier counter
    float*  pHalo  = (float*)d_ws;
    float*  vzHalo = pHalo + (size_t)NBLK * NX_;
    uint32* bar    = (uint32*)((char*)d_ws + (size_t)2 * NBLK * NX_ * sizeof(float));

    (void)hipMemsetAsync(bar, 0, 64, stream);    // reset barrier counter (capture-safe)

    hipLaunchKernelGGL(wave_persistent, dim3(NBLK), dim3(NX_), 0, stream,
                       wav, vp, rho, Q, srcZ, srcX, recX, recZ,
                       out, pHalo, vzHalo, bar);
}